// GNN_54597624267027
// MI455X (gfx1250) — compile-verified
//
#include <hip/hip_runtime.h>

typedef __attribute__((ext_vector_type(2))) float v2f;
typedef __attribute__((ext_vector_type(8))) float v8f;

#define KMAX 128  // K is 128 for every layer in this model

// ---------------------------------------------------------------------------
// GEMM via V_WMMA_F32_16X16X4_F32:  out[N x Mw] = A[N x K] @ W[K x Mw]
// Mw is a multiple of 16 (layer 3 weights are zero-padded 40 -> 48).
// Block = 128 threads (4 waves) computing a 64x16 output strip (4 row-tiles,
// one column-tile). The K x 16 B-panel is staged once in LDS as interleaved
// (k, k+1) pairs so each lane's B fragment is a single ds_load_b64.
// Fragment layouts (CDNA5 ISA):
//   A 16x4 f32: lane l -> row (l&15), K = (l>>4)*2 .. +1
//   B  4x16 f32: lane l -> col (l&15), K = (l>>4)*2 .. +1
//   D: VGPR j -> row (l>>4)*8 + j, col (l&15)
// ---------------------------------------------------------------------------
__global__ __launch_bounds__(128) void gcn_gemm_wmma_f32(
    const float* __restrict__ A, const float* __restrict__ W,
    float* __restrict__ out, int Nrows, int K, int Mw,
    int tiles_n, int tiles_m) {
  __shared__ float Wl[KMAX * 16];  // pair-interleaved: (k,n) at ((k>>1)*16+n)*2+(k&1)

  const int tid  = threadIdx.x;
  const int wave = tid >> 5;
  const int lane = tid & 31;
  const int tn   = blockIdx.x % tiles_n;
  const int tm   = (blockIdx.x / tiles_n) * 4 + wave;
  const int n0   = tn * 16;

  // ---- stage B panel (all 128 threads, coalesced global reads) ----
  for (int idx = tid; idx < K * 16; idx += 128) {
    int k = idx >> 4, n = idx & 15;
    Wl[(((k >> 1) << 4) + n) * 2 + (k & 1)] = W[(size_t)k * Mw + n0 + n];
  }
  __syncthreads();

  if (tm >= tiles_m) return;  // wave-uniform (EXEC stays all-1s for WMMA)

  const int mrow = tm * 16 + (lane & 15);
  const int kb   = (lane >> 4) * 2;  // 0 or 2
  const float* __restrict__ Arow = A + (size_t)mrow * K;
  const v2f* __restrict__ Wp = (const v2f*)Wl;  // pair (k,k+1) at [ (k>>1)*16 + n ]

  v8f acc = {};
#pragma unroll 4
  for (int k0 = 0; k0 < K; k0 += 4) {
    int k = k0 + kb;
    v2f a = *(const v2f*)(Arow + k);          // global_load_b64, contiguous
    v2f b = Wp[((k >> 1) << 4) + (lane & 15)];  // ds_load_b64
    acc = __builtin_amdgcn_wmma_f32_16x16x4_f32(
        false, a, false, b, (short)0, acc, false, false);
  }

  const int rbase = tm * 16 + (lane >> 4) * 8;
  const int ncol  = n0 + (lane & 15);
#pragma unroll
  for (int j = 0; j < 8; ++j)
    out[(size_t)(rbase + j) * Mw + ncol] = acc[j];
}

// ---------------------------------------------------------------------------
// Zero-pad W3 [K x C] into W3p [K x Cp]
// ---------------------------------------------------------------------------
__global__ void gcn_pad_w(const float* __restrict__ Wsrc, float* __restrict__ Wpad,
                          int K, int C, int Cp) {
  int idx = blockIdx.x * blockDim.x + threadIdx.x;
  if (idx >= K * Cp) return;
  int k = idx / Cp, n = idx - k * Cp;
  Wpad[idx] = (n < C) ? Wsrc[k * C + n] : 0.0f;
}

// ---------------------------------------------------------------------------
// Graph-aggregation helpers
// ---------------------------------------------------------------------------
__global__ void gcn_fill_f32(float* __restrict__ p, long long n, float v) {
  long long i = (long long)blockIdx.x * blockDim.x + threadIdx.x;
  if (i < n) p[i] = v;
}

__global__ void gcn_degree(const int* __restrict__ dst, float* __restrict__ deg,
                           long long E) {
  long long e = (long long)blockIdx.x * blockDim.x + threadIdx.x;
  if (e < E) atomicAdd(&deg[dst[e]], 1.0f);
}

__global__ void gcn_rsqrt_inplace(float* __restrict__ deg, int N) {
  int i = blockIdx.x * blockDim.x + threadIdx.x;
  if (i < N) deg[i] = rsqrtf(deg[i]);
}

// One wave per edge, H == 128: lane covers 4 contiguous features (float4 load,
// four global_atomic_add_f32 that resolve in L2 -- agg fits in the 192MB L2).
__global__ __launch_bounds__(256) void gcn_scatter128(
    const float* __restrict__ h, const int* __restrict__ src,
    const int* __restrict__ dst, const float* __restrict__ dinv,
    float* __restrict__ agg, long long E) {
  long long tid = (long long)blockIdx.x * blockDim.x + threadIdx.x;
  long long e = tid >> 5;
  int lane = threadIdx.x & 31;
  if (e >= E) return;
  int s = src[e], d = dst[e];
  float c = dinv[s] * dinv[d];
  const float4* hs = (const float4*)(h + (size_t)s * 128);
  float4 v = hs[lane];
  float* base = agg + (size_t)d * 128 + lane * 4;
  atomicAdd(base + 0, v.x * c);
  atomicAdd(base + 1, v.y * c);
  atomicAdd(base + 2, v.z * c);
  atomicAdd(base + 3, v.w * c);
}

// One wave per edge, small feature dim; h has row-stride hstride (padded).
__global__ __launch_bounds__(256) void gcn_scatter_small(
    const float* __restrict__ h, const int* __restrict__ src,
    const int* __restrict__ dst, const float* __restrict__ dinv,
    float* __restrict__ agg, long long E, int C, int hstride) {
  long long tid = (long long)blockIdx.x * blockDim.x + threadIdx.x;
  long long e = tid >> 5;
  int lane = threadIdx.x & 31;
  if (e >= E) return;
  int s = src[e], d = dst[e];
  float c = dinv[s] * dinv[d];
  for (int f = lane; f < C; f += 32)
    atomicAdd(&agg[(size_t)d * C + f], h[(size_t)s * hstride + f] * c);
}

// agg = (relu?) (agg + h * dinv^2 + bias); h has row-stride hstride
__global__ void gcn_finalize(const float* __restrict__ h,
                             const float* __restrict__ dinv,
                             const float* __restrict__ bias,
                             float* __restrict__ agg, long long NC, int C,
                             int hstride, int do_relu) {
  long long idx = (long long)blockIdx.x * blockDim.x + threadIdx.x;
  if (idx >= NC) return;
  long long i = idx / C;
  int f = (int)(idx - i * C);
  float di = dinv[i];
  float v = agg[idx] + h[i * hstride + f] * di * di + bias[f];
  if (do_relu) v = fmaxf(v, 0.0f);
  agg[idx] = v;
}

// ---------------------------------------------------------------------------
// Orchestration
// ---------------------------------------------------------------------------
static inline size_t align_up(size_t x) { return (x + 255) & ~(size_t)255; }

extern "C" void kernel_launch(void* const* d_in, const int* in_sizes, int n_in,
                              void* d_out, int out_size, void* d_ws,
                              size_t ws_size, hipStream_t stream) {
  const float* x  = (const float*)d_in[0];
  const int*   ei = (const int*)d_in[1];
  const float* W1 = (const float*)d_in[2];
  const float* b1 = (const float*)d_in[3];
  const float* W2 = (const float*)d_in[4];
  const float* b2 = (const float*)d_in[5];
  const float* W3 = (const float*)d_in[6];
  const float* b3 = (const float*)d_in[7];
  float* out = (float*)d_out;

  const int H = in_sizes[3];               // 128
  const int C = in_sizes[7];               // 40
  const int F = in_sizes[2] / H;           // 128
  const int N = in_sizes[0] / F;           // 50000
  const long long E = (long long)in_sizes[1] / 2;
  const int Cp = ((C + 15) / 16) * 16;     // 48

  const int* src = ei;
  const int* dst = ei + E;

  // Workspace carve-up (buffers reused across layers; stream order makes it safe)
  char* ws = (char*)d_ws;
  size_t off = 0;
  float* deg  = (float*)(ws + off); off += align_up((size_t)N * 4);            // -> dinv
  float* hbuf = (float*)(ws + off); off += align_up((size_t)N * H * 4);        // gemm out
  float* abuf = (float*)(ws + off); off += align_up((size_t)N * H * 4);        // agg / act
  float* h3p  = (float*)(ws + off); off += align_up((size_t)N * Cp * 4);       // layer3 gemm (padded)
  float* W3p  = (float*)(ws + off); off += align_up((size_t)H * Cp * 4);       // padded W3

  const int T = 256;
  const long long NH = (long long)N * H;
  const long long NC = (long long)N * C;
  const int tiles_m = (N + 15) / 16;
  const int grid_m  = (tiles_m + 3) / 4;   // 4 row-tiles per block

  // --- degrees: deg = 1 + in-degree, then dinv = rsqrt(deg) -----------------
  gcn_fill_f32<<<(N + T - 1) / T, T, 0, stream>>>(deg, N, 1.0f);
  gcn_degree<<<(int)((E + T - 1) / T), T, 0, stream>>>(dst, deg, E);
  gcn_rsqrt_inplace<<<(N + T - 1) / T, T, 0, stream>>>(deg, N);

  // --- layer 1: h = x@W1 ; agg ; relu(agg + h/deg + b1) ---------------------
  {
    int tiles_n = H / 16;
    gcn_gemm_wmma_f32<<<grid_m * tiles_n, 128, 0, stream>>>(x, W1, hbuf, N, F, H,
                                                            tiles_n, tiles_m);
    gcn_fill_f32<<<(int)((NH + T - 1) / T), T, 0, stream>>>(abuf, NH, 0.0f);
    gcn_scatter128<<<(int)((E * 32 + T - 1) / T), T, 0, stream>>>(hbuf, src, dst,
                                                                  deg, abuf, E);
    gcn_finalize<<<(int)((NH + T - 1) / T), T, 0, stream>>>(hbuf, deg, b1, abuf,
                                                            NH, H, H, 1);
  }

  // --- layer 2 --------------------------------------------------------------
  {
    int tiles_n = H / 16;
    gcn_gemm_wmma_f32<<<grid_m * tiles_n, 128, 0, stream>>>(abuf, W2, hbuf, N, H,
                                                            H, tiles_n, tiles_m);
    gcn_fill_f32<<<(int)((NH + T - 1) / T), T, 0, stream>>>(abuf, NH, 0.0f);
    gcn_scatter128<<<(int)((E * 32 + T - 1) / T), T, 0, stream>>>(hbuf, src, dst,
                                                                  deg, abuf, E);
    gcn_finalize<<<(int)((NH + T - 1) / T), T, 0, stream>>>(hbuf, deg, b2, abuf,
                                                            NH, H, H, 1);
  }

  // --- layer 3 (C = 40 padded to 48, no relu), accumulate into d_out --------
  {
    int tiles_n = Cp / 16;
    gcn_pad_w<<<(H * Cp + T - 1) / T, T, 0, stream>>>(W3, W3p, H, C, Cp);
    gcn_gemm_wmma_f32<<<grid_m * tiles_n, 128, 0, stream>>>(abuf, W3p, h3p, N, H,
                                                            Cp, tiles_n, tiles_m);
    gcn_fill_f32<<<(int)((NC + T - 1) / T), T, 0, stream>>>(out, NC, 0.0f);
    gcn_scatter_small<<<(int)((E * 32 + T - 1) / T), T, 0, stream>>>(
        h3p, src, dst, deg, out, E, C, Cp);
    gcn_finalize<<<(int)((NC + T - 1) / T), T, 0, stream>>>(h3p, deg, b3, out,
                                                            NC, C, Cp, 0);
  }
}